// VQVAE_2027224564517
// MI455X (gfx1250) — compile-verified
//
#include <hip/hip_runtime.h>
#include <math.h>

typedef __attribute__((ext_vector_type(2))) float v2f;
typedef __attribute__((ext_vector_type(8))) float v8f;

#define LRELU_SLOPE 0.01f

// ---------------------------------------------------------------------------
// Implicit-GEMM conv kernel using fp32 WMMA (V_WMMA_F32_16X16X4_F32).
//   LAYER 0: conv 4x4 stride 2 pad 1 (SAME)       -- 16 taps
//   LAYER 1: 1x1 conv (pure GEMM)                 -- 1 tap
//   LAYER 2: conv-transpose 4x4 s2 SAME, split into 4 parity classes
//            (blockIdx.z); each class is a dense 2x2-tap conv -- 4 taps
// Each wave computes a 32(M) x 16*NT(N) tile: per k-group (4 K values) it
// issues 2 A-fragment b64 loads + NT B-fragment b64 loads and 2*NT WMMAs.
// Tap base addresses / bounds checks are hoisted out of the channel loop.
// Weights come pre-packed as Bpack[g][n][4] (zeros baked into padded slots).
// ---------------------------------------------------------------------------
template<int LAYER, int BATCH, int IH, int IW, int CIN,
         int OH, int OW, int COUT, int NT, int ACT>
__global__ __launch_bounds__(256)
void igemm_wmma(const float* __restrict__ in, const float* __restrict__ bpack,
                const float* __restrict__ bias, float* __restrict__ out)
{
    constexpr int CINP  = (CIN + 3) & ~3;
    constexpr int TAPS  = (LAYER == 0) ? 16 : (LAYER == 1 ? 1 : 4);
    constexpr int COUTP = (COUT + 15) & ~15;
    constexpr int OHE   = (LAYER == 2) ? OH / 2 : OH;   // per-parity-class dims
    constexpr int OWE   = (LAYER == 2) ? OW / 2 : OW;
    constexpr int M     = BATCH * OHE * OWE;
    constexpr int MT    = M / 32;
    constexpr int NTN   = COUTP / (16 * NT);
    constexpr int PACKSZ = TAPS * CINP * COUTP;

    const int ph = (LAYER == 2) ? (int)(blockIdx.z >> 1) : 0;
    const int pw = (LAYER == 2) ? (int)(blockIdx.z & 1) : 0;
    const float* __restrict__ bp = bpack + (LAYER == 2 ? (int)blockIdx.z * PACKSZ : 0);

    const int lane = threadIdx.x & 31;
    const int wid  = (blockIdx.x * 256 + threadIdx.x) >> 5;
    const int mt   = wid / NTN;
    const int nt   = wid % NTN;
    if (mt >= MT) return;                 // wave-uniform (EXEC stays all-ones)

    const int hi = lane >> 4;
    const int r  = lane & 15;

    // Two row-blocks of 16 rows handled by this wave; per-lane row coords.
    int bR[2], ohR[2], owR[2];
    #pragma unroll
    for (int rb = 0; rb < 2; ++rb) {
        const int mA  = mt * 32 + rb * 16 + r;
        bR[rb]        = mA / (OHE * OWE);
        const int rem = mA % (OHE * OWE);
        int oh = rem / OWE, ow = rem % OWE;
        if (LAYER == 2) { oh = oh * 2 + ph; ow = ow * 2 + pw; }
        ohR[rb] = oh; owR[rb] = ow;
    }

    v8f acc[2][NT] = {};
    const v2f zz = {0.0f, 0.0f};

    #pragma unroll
    for (int tap = 0; tap < TAPS; ++tap) {
        // ---- per-tap base address + validity (hoisted out of channel loop) ----
        int base[2]; bool val[2];
        #pragma unroll
        for (int rb = 0; rb < 2; ++rb) {
            if (LAYER == 1) {
                base[rb] = (mt * 32 + rb * 16 + r) * CIN;
                val[rb]  = true;
            } else if (LAYER == 0) {
                const int kh = tap >> 2, kw = tap & 3;
                const int ih = ohR[rb] * 2 + kh - 1;
                const int iw = owR[rb] * 2 + kw - 1;
                val[rb]  = (ih >= 0 && ih < IH && iw >= 0 && iw < IW);
                base[rb] = ((bR[rb] * IH + ih) * IW + iw) * CIN;
            } else {
                const int th = tap >> 1, tw = tap & 1;
                const int kh = 2 * th + ph, kw = 2 * tw + pw;
                const int ihn = ohR[rb] + kh - 2;   // even by construction
                const int iwn = owR[rb] + kw - 2;
                const int ih = ihn >> 1, iw = iwn >> 1;
                val[rb]  = (ihn >= 0 && ih < IH && iwn >= 0 && iw < IW);
                base[rb] = ((bR[rb] * IH + ih) * IW + iw) * CIN;
            }
        }
        // ---- channel loop: contiguous, vectorized A loads ----
        #pragma unroll 4
        for (int ci0 = 0; ci0 < CINP; ci0 += 4) {
            v2f a[2];
            #pragma unroll
            for (int rb = 0; rb < 2; ++rb) {
                if constexpr (CINP == CIN) {
                    a[rb] = val[rb] ? *(const v2f*)&in[base[rb] + ci0 + hi * 2] : zz;
                } else {            // conv1: CIN=3 padded to 4, scalar fallback
                    v2f t = zz;
                    #pragma unroll
                    for (int e = 0; e < 2; ++e) {
                        const int ci = ci0 + hi * 2 + e;
                        float v = 0.0f;
                        if (val[rb] && ci < CIN) v = in[base[rb] + ci];
                        if (e == 0) t.x = v; else t.y = v;
                    }
                    a[rb] = t;
                }
            }
            const int g = tap * (CINP / 4) + (ci0 >> 2);
            #pragma unroll
            for (int j = 0; j < NT; ++j) {
                const int n  = nt * (16 * NT) + j * 16 + r;
                const v2f bf = *(const v2f*)&bp[(g * COUTP + n) * 4 + hi * 2];
                acc[0][j] = __builtin_amdgcn_wmma_f32_16x16x4_f32(
                                false, a[0], false, bf, (short)0, acc[0][j], false, false);
                acc[1][j] = __builtin_amdgcn_wmma_f32_16x16x4_f32(
                                false, a[1], false, bf, (short)0, acc[1][j], false, false);
            }
        }
    }

    // ---- Epilogue: bias + activation + store ----
    #pragma unroll
    for (int rb = 0; rb < 2; ++rb) {
        #pragma unroll
        for (int j = 0; j < NT; ++j) {
            const int col = nt * (16 * NT) + j * 16 + r;
            if (col >= COUT) continue;
            const float bv = bias[col];
            #pragma unroll
            for (int vg = 0; vg < 8; ++vg) {
                const int m2   = mt * 32 + rb * 16 + vg + hi * 8;  // C: M = vg + 8*half
                const int b2   = m2 / (OHE * OWE);
                const int rem2 = m2 % (OHE * OWE);
                const int oh2  = rem2 / OWE;
                const int ow2  = rem2 % OWE;
                float v = acc[rb][j][vg] + bv;
                if (ACT == 1)      v = (v > 0.0f) ? v : LRELU_SLOPE * v;
                else if (ACT == 2) v = tanhf(v);
                if (LAYER == 2) {
                    const int oh = oh2 * 2 + ph, ow = ow2 * 2 + pw;
                    out[((b2 * OH + oh) * OW + ow) * COUT + col] = v;
                } else {
                    out[m2 * COUT + col] = v;
                }
            }
        }
    }
}

// ---------------------------------------------------------------------------
// Weight packing: Bpack[g][n][4] with g = tap*(CINP/4)+ci0/4, so every lane's
// B fragment is one contiguous b64 load. Zeros baked into padded slots.
// ---------------------------------------------------------------------------
__global__ void pack_conv_kernel(const float* __restrict__ w, float* __restrict__ bpk,
                                 int taps, int cin, int cinp, int cout, int coutp)
{
    const int gid = blockIdx.x * 256 + threadIdx.x;
    const int total = taps * cinp * coutp;
    if (gid >= total) return;
    const int j   = gid & 3;
    const int n   = (gid >> 2) % coutp;
    const int g   = (gid >> 2) / coutp;
    const int kp  = g * 4 + j;
    const int ci  = kp % cinp;
    const int tap = kp / cinp;
    float v = 0.0f;
    if (ci < cin && n < cout) v = w[(tap * cin + ci) * cout + n];
    bpk[gid] = v;
}

__global__ void pack_convT_kernel(const float* __restrict__ w, float* __restrict__ bpk,
                                  int cin, int cout, int coutp)
{
    const int par = blockIdx.y;             // parity class (ph,pw)
    const int ph = par >> 1, pw = par & 1;
    const int gid = blockIdx.x * 256 + threadIdx.x;
    const int total = 4 * cin * coutp;      // 4 taps, cin divisible by 4 here
    if (gid >= total) return;
    const int j   = gid & 3;
    const int n   = (gid >> 2) % coutp;
    const int g   = (gid >> 2) / coutp;
    const int kp  = g * 4 + j;
    const int ci  = kp % cin;
    const int tap = kp / cin;
    const int th = tap >> 1, tw = tap & 1;
    const int kh = 2 * th + ph, kw = 2 * tw + pw;
    float v = 0.0f;
    if (n < cout) v = w[((kh * 4 + kw) * cin + ci) * cout + n];
    bpk[par * total + gid] = v;
}

// ---------------------------------------------------------------------------
// VQ: codebook squared norms
// ---------------------------------------------------------------------------
__global__ void enorm_kernel(const float* __restrict__ emb, float* __restrict__ enorm)
{
    const int k = blockIdx.x * blockDim.x + threadIdx.x;
    if (k >= 2048) return;
    float s = 0.f;
    #pragma unroll
    for (int j = 0; j < 64; ++j) { const float e = emb[k * 64 + j]; s = fmaf(e, e, s); }
    enorm[k] = s;
}

// ---------------------------------------------------------------------------
// VQ argmin: each thread owns one latent vector (over the H' axis, matching
// the reference's transpose(0,2,3,1)); codebook streamed through LDS in
// 128-code (32KB) chunks. Strict '<' keeps the first minimum (jnp.argmin).
// ---------------------------------------------------------------------------
__global__ __launch_bounds__(256)
void vq_argmin_kernel(const float* __restrict__ enc, const float* __restrict__ emb,
                      const float* __restrict__ enorm, int* __restrict__ idx)
{
    __shared__ float smem[128 * 64];
    __shared__ float snorm[128];
    const int t = threadIdx.x;
    const int n = blockIdx.x * 256 + t;          // n = (b*64 + w)*64 + d
    const int d = n & 63;
    const int w = (n >> 6) & 63;
    const int b = n >> 12;

    float x[64];
    #pragma unroll
    for (int h = 0; h < 64; ++h)
        x[h] = enc[((b * 64 + h) * 64 + w) * 64 + d];  // enc[b,h,w,d], vector over h
    float xn = 0.f;
    #pragma unroll
    for (int h = 0; h < 64; ++h) xn = fmaf(x[h], x[h], xn);

    float best = 3.4e38f;
    int bestk = 0;
    for (int c = 0; c < 16; ++c) {
        __syncthreads();
        for (int i = t; i < 128 * 64; i += 256) smem[i] = emb[c * 128 * 64 + i];
        if (t < 128) snorm[t] = enorm[c * 128 + t];
        __syncthreads();
        for (int kk = 0; kk < 128; ++kk) {
            float dot = 0.f;
            #pragma unroll
            for (int h = 0; h < 64; ++h) dot = fmaf(x[h], smem[kk * 64 + h], dot);
            const float dist = xn + snorm[kk] - 2.0f * dot;
            const int kidx = c * 128 + kk;
            if (dist < best) { best = dist; bestk = kidx; }
        }
    }
    idx[n] = bestk;
}

// ---------------------------------------------------------------------------
// Gather quantized vectors (transposed back to [B,H,W,D] via indexing) and
// accumulate sum((q - z)^2) for the vq loss.
// ---------------------------------------------------------------------------
__global__ __launch_bounds__(256)
void vq_gather_loss_kernel(const float* __restrict__ enc, const float* __restrict__ emb,
                           const int* __restrict__ idx, float* __restrict__ quant,
                           float* __restrict__ loss_acc)
{
    __shared__ float red[256];
    const int tid = blockIdx.x * 256 + threadIdx.x;   // = ((b*64+h)*64+w)*64+d
    const int d = tid & 63;
    const int w = (tid >> 6) & 63;
    const int h = (tid >> 12) & 63;
    const int b = tid >> 18;
    const int k = idx[(b << 12) + (w << 6) + d];
    const float q = emb[k * 64 + h];
    const float e = enc[tid];
    quant[tid] = q;                                   // straight-through forward value
    const float diff = q - e;
    red[threadIdx.x] = diff * diff;
    __syncthreads();
    for (int s = 128; s > 0; s >>= 1) {
        if (threadIdx.x < s) red[threadIdx.x] += red[threadIdx.x + s];
        __syncthreads();
    }
    if (threadIdx.x == 0) atomicAdd(loss_acc, red[0]);
}

__global__ void loss_finalize_kernel(const float* __restrict__ loss_acc,
                                     float* __restrict__ out_loss)
{
    // vq_loss = embedding_loss + 0.25*commitment_loss = 1.25 * mean(diff^2)
    out_loss[0] = 1.25f * loss_acc[0] / 4194304.0f;   // 16*64*64*64 elements
}

// ---------------------------------------------------------------------------
extern "C" void kernel_launch(void* const* d_in, const int* in_sizes, int n_in,
                              void* d_out, int out_size, void* d_ws, size_t ws_size,
                              hipStream_t stream)
{
    (void)in_sizes; (void)n_in; (void)out_size; (void)ws_size;
    const float* x   = (const float*)d_in[0];
    const float* emb = (const float*)d_in[1];
    const float* w1  = (const float*)d_in[2];
    const float* b1  = (const float*)d_in[3];
    const float* w2  = (const float*)d_in[4];
    const float* b2  = (const float*)d_in[5];
    const float* w3  = (const float*)d_in[6];
    const float* b3  = (const float*)d_in[7];
    const float* dw1 = (const float*)d_in[8];
    const float* db1 = (const float*)d_in[9];
    const float* dw2 = (const float*)d_in[10];
    const float* db2 = (const float*)d_in[11];

    char* ws = (char*)d_ws;
    float* H1v   = (float*)(ws + 0);                     // [16,128,128,128] 134.2MB
    float* H2v   = (float*)(ws + 134217728ULL);          // [16,64,64,256]    67.1MB
    float* ENC   = (float*)(ws + 201326592ULL);          // [16,64,64,64]     16.8MB
    int*   IDX   = (int*)  (ws + 218103808ULL);          // [65536]
    float* ENORM = (float*)(ws + 218365952ULL);          // [2048]
    float* G     = (float*)(ws + 0);                     // [16,128,128,256] 268.4MB (reuses encoder region)
    float* QUANT = (float*)(ws + 268435456ULL);          // [16,64,64,64]     16.8MB
    float* LOSS  = (float*)(ws + 285212672ULL);          // scalar accumulator
    float* PK1   = (float*)(ws + 285213184ULL);          // 16*4*128      =   32KB
    float* PK2   = (float*)(ws + 285245952ULL);          // 16*128*256    =    2MB
    float* PK3   = (float*)(ws + 287343104ULL);          // 1*256*64      =   64KB
    float* PKT1  = (float*)(ws + 287408640ULL);          // 4par*4*64*256 =    1MB
    float* PKT2  = (float*)(ws + 288457216ULL);          // 4par*4*256*16 =  256KB
    float* recon    = (float*)d_out;                     // [16,256,256,3]
    float* out_loss = recon + 3145728;

    // ---- pack weights into fragment-ready layout ----
    pack_conv_kernel<<<(16 * 4 * 128 + 255) / 256, 256, 0, stream>>>(w1, PK1, 16, 3, 4, 128, 128);
    pack_conv_kernel<<<(16 * 128 * 256) / 256, 256, 0, stream>>>(w2, PK2, 16, 128, 128, 256, 256);
    pack_conv_kernel<<<(1 * 256 * 64) / 256, 256, 0, stream>>>(w3, PK3, 1, 256, 256, 64, 64);
    pack_convT_kernel<<<dim3((4 * 64 * 256) / 256, 4, 1), 256, 0, stream>>>(dw1, PKT1, 64, 256, 256);
    pack_convT_kernel<<<dim3((4 * 256 * 16) / 256, 4, 1), 256, 0, stream>>>(dw2, PKT2, 256, 3, 16);

    // conv1: [16,256,256,3] -> [16,128,128,128], leaky. M=262144 -> MT=8192, NTN=2
    igemm_wmma<0, 16, 256, 256, 3, 128, 128, 128, 4, 1>
        <<<dim3(8192 * 2 / 8, 1, 1), 256, 0, stream>>>(x, PK1, b1, H1v);
    // conv2: [16,128,128,128] -> [16,64,64,256], leaky. MT=2048, NTN=4
    igemm_wmma<0, 16, 128, 128, 128, 64, 64, 256, 4, 1>
        <<<dim3(2048 * 4 / 8, 1, 1), 256, 0, stream>>>(H1v, PK2, b2, H2v);
    // conv3 (1x1): [16,64,64,256] -> [16,64,64,64]. MT=2048, NTN=1
    igemm_wmma<1, 16, 64, 64, 256, 64, 64, 64, 4, 0>
        <<<dim3(2048 * 1 / 8, 1, 1), 256, 0, stream>>>(H2v, PK3, b3, ENC);

    // ---- vector quantizer ----
    enorm_kernel<<<8, 256, 0, stream>>>(emb, ENORM);
    vq_argmin_kernel<<<256, 256, 0, stream>>>(ENC, emb, ENORM, IDX);
    hipMemsetAsync(LOSS, 0, sizeof(float), stream);
    vq_gather_loss_kernel<<<16384, 256, 0, stream>>>(ENC, emb, IDX, QUANT, LOSS);
    loss_finalize_kernel<<<1, 1, 0, stream>>>(LOSS, out_loss);

    // convT1: [16,64,64,64] -> [16,128,128,256], leaky. per-class MT=2048, NTN=4
    igemm_wmma<2, 16, 64, 64, 64, 128, 128, 256, 4, 1>
        <<<dim3(2048 * 4 / 8, 1, 4), 256, 0, stream>>>(QUANT, PKT1, db1, G);
    // convT2: [16,128,128,256] -> [16,256,256,3], tanh. per-class MT=8192, NTN=1
    igemm_wmma<2, 16, 128, 128, 256, 256, 256, 3, 1, 2>
        <<<dim3(8192 * 1 / 8, 1, 4), 256, 0, stream>>>(G, PKT2, db2, recon);
}